// DistMultDecoder_34041910788102
// MI455X (gfx1250) — compile-verified
//
#include <hip/hip_runtime.h>

#define HIDDEN 128
#define LDS_STRIDE 132   // 128 + 4 pad: wave's 64 dword B-reads hit all 64 banks once

typedef __attribute__((ext_vector_type(2))) float v2f;
typedef __attribute__((ext_vector_type(8))) float v8f;

// ---------------------------------------------------------------------------
// Kernel 1: Wsym = W + W^T   (128 x 128)
// ---------------------------------------------------------------------------
__global__ void wsym_kernel(const float* __restrict__ W, float* __restrict__ Wsym) {
    int i = blockIdx.x;    // row
    int j = threadIdx.x;   // col
    Wsym[i * HIDDEN + j] = W[i * HIDDEN + j] + W[j * HIDDEN + i];
}

// ---------------------------------------------------------------------------
// Kernel 2: Y = Z @ Wsym   (nrows x 128) @ (128 x 128), fp32 WMMA
// One wave computes a 16-row x 128-col output strip; Wsym staged in LDS.
// ---------------------------------------------------------------------------
__global__ void __launch_bounds__(256)
gemm_kernel(const float* __restrict__ Z,
            const float* __restrict__ Wsym,
            float* __restrict__ Y, int nrows) {
    __shared__ float wsh[HIDDEN * LDS_STRIDE];   // ~66 KB of the 320 KB WGP LDS

    // --- cooperative stage of Wsym into LDS (float4, 16B-aligned rows) ------
    {
        int tid = threadIdx.x;
#pragma unroll
        for (int it = 0; it < (HIDDEN * HIDDEN / 4) / 256; ++it) {  // 16 iters
            int idx4 = tid + it * 256;       // float4 index
            int row  = idx4 >> 5;            // 32 float4 per row
            int c4   = idx4 & 31;
            float4 v = ((const float4*)Wsym)[idx4];
            ((float4*)(wsh + row * LDS_STRIDE))[c4] = v;
        }
    }
    __syncthreads();   // all waves participate (no early return before here)

    int wave = (blockIdx.x * blockDim.x + threadIdx.x) >> 5;
    int lane = threadIdx.x & 31;
    int m0   = wave * 16;
    bool active = (m0 < nrows);              // wave-uniform
    if (!active) return;                     // EXEC all-1s for WMMA below

    int half = lane >> 4;                    // 0: lanes 0-15, 1: lanes 16-31
    int lid  = lane & 15;

    v8f acc[8];
#pragma unroll
    for (int n = 0; n < 8; ++n) acc[n] = v8f{};

    int arow_idx = m0 + lid;
    if (arow_idx >= nrows) arow_idx = nrows - 1;   // safety clamp
    const float* arow = Z + (size_t)arow_idx * HIDDEN + half * 2;

    // K loop in two stages of 16 chunks; A fragments preloaded per stage so
    // global loads (LOADcnt) overlap with LDS reads (DScnt) + WMMA.
    // Fully unrolled so areg[] indexing is static (no v_movrels/m0 indexing).
#pragma unroll
    for (int stage = 0; stage < 2; ++stage) {
        v2f areg[16];
#pragma unroll
        for (int i = 0; i < 16; ++i)
            areg[i] = *(const v2f*)(arow + (stage * 16 + i) * 4);

#pragma unroll
        for (int kc = 0; kc < 16; ++kc) {
            int ka = (stage * 16 + kc) * 4 + half * 2;
            v2f a = areg[kc];
            v2f b[8];
#pragma unroll
            for (int n = 0; n < 8; ++n)     // conflict-free ds_load_b64 batch
                b[n] = *(const v2f*)(wsh + (n * 16 + lid) * LDS_STRIDE + ka);
#pragma unroll
            for (int n = 0; n < 8; ++n)
                acc[n] = __builtin_amdgcn_wmma_f32_16x16x4_f32(
                    false, a, false, b[n], (short)0, acc[n], false, false);
        }
    }

    // C/D layout: VGPR r -> M = r (lanes 0-15) / M = 8 + r (lanes 16-31), N = lid
#pragma unroll
    for (int n = 0; n < 8; ++n) {
        int n0 = n * 16;
#pragma unroll
        for (int r = 0; r < 8; ++r) {
            int m = m0 + half * 8 + r;
            if (m < nrows)
                Y[(size_t)m * HIDDEN + n0 + lid] = acc[n][r];
        }
    }
}

// ---------------------------------------------------------------------------
// Kernel 3: per-edge score = sigmoid( dot(Y[src], Z[dst]) )
// One wave per edge: 32 lanes x float4 covers the 128-wide row.
// ---------------------------------------------------------------------------
__global__ void edge_kernel(const long long* __restrict__ ei,  // [2, E] int64
                            const float* __restrict__ Y,
                            const float* __restrict__ Z,
                            float* __restrict__ out, int E) {
    int wave = blockIdx.x * (blockDim.x >> 5) + (threadIdx.x >> 5);
    int lane = threadIdx.x & 31;
    if (wave >= E) return;

    // Streaming, read-once data: non-temporal so Y/Z stay resident in L2
    long long s = __builtin_nontemporal_load(ei + wave);      // src row
    long long d = __builtin_nontemporal_load(ei + E + wave);  // dst row

    const float4* ys = (const float4*)(Y + (size_t)s * HIDDEN);
    const float4* zd = (const float4*)(Z + (size_t)d * HIDDEN);
    float4 a = ys[lane];      // coalesced 512B row gather (L2 resident)
    float4 b = zd[lane];

    float p = a.x * b.x + a.y * b.y + a.z * b.z + a.w * b.w;

    // wave32 butterfly reduction
#pragma unroll
    for (int off = 16; off >= 1; off >>= 1)
        p += __shfl_xor(p, off, 32);

    if (lane == 0) {
        float sig = 1.0f / (1.0f + __expf(-p));
        __builtin_nontemporal_store(sig, out + wave);
    }
}

// ---------------------------------------------------------------------------
extern "C" void kernel_launch(void* const* d_in, const int* in_sizes, int n_in,
                              void* d_out, int out_size, void* d_ws, size_t ws_size,
                              hipStream_t stream) {
    const float*     z  = (const float*)d_in[0];      // [N, 128] fp32
    const long long* ei = (const long long*)d_in[1];  // [2, E] int64
    const float*     w  = (const float*)d_in[2];      // [128, 128] fp32
    float*           out = (float*)d_out;

    int nnodes = in_sizes[0] / HIDDEN;
    int E      = in_sizes[1] / 2;

    float* wsym = (float*)d_ws;                                            // 64 KB
    float* Y    = (float*)((char*)d_ws + HIDDEN * HIDDEN * sizeof(float)); // 51.2 MB

    // 1) symmetrize W
    wsym_kernel<<<HIDDEN, HIDDEN, 0, stream>>>(w, wsym);

    // 2) Y = Z @ Wsym  (one wave per 16-row strip, 8 waves / 256-thread block)
    int strips  = (nnodes + 15) / 16;
    int gblocks = (strips + 7) / 8;
    gemm_kernel<<<gblocks, 256, 0, stream>>>(z, wsym, Y, nnodes);

    // 3) per-edge bilinear score (one wave per edge, 8 edges / block)
    int eblocks = (E + 7) / 8;
    edge_kernel<<<eblocks, 256, 0, stream>>>(ei, Y, z, out, E);
}